// OptimizedMoELayer_18184891532045
// MI455X (gfx1250) — compile-verified
//
#include <hip/hip_runtime.h>
#include <hip/hip_bf16.h>

typedef __attribute__((ext_vector_type(2))) float    v2f;
typedef __attribute__((ext_vector_type(8))) float    v8f;
typedef __attribute__((ext_vector_type(4))) unsigned u32x4;
typedef __attribute__((ext_vector_type(4))) int      i32x4;
typedef __attribute__((ext_vector_type(8))) int      i32x8;

#define T_TOK   2048   // B*S
#define D_DIM   512
#define F_DIM   2048
#define E_EXP   16
#define TILE_F  128    // F columns per block (8 waves x 16)
// TDM pad: +2 dwords after every 256 stored dwords -> effective LDS row stride
// for a 512-dword row is 516 dwords. 516 % 64 == 4, so the 16 lanes of a WMMA
// A/B fetch (row n -> offset 516*n) hit 16 distinct LDS banks.
#define W1_STRIDE 516
#define SEG_STRIDE 258  // 256 data dwords + 2 pad dwords

// ---------------------------------------------------------------------------
// Tensor Data Mover descriptor helpers (CDNA5 D#, CDNA5 ISA)
// ---------------------------------------------------------------------------
__device__ __forceinline__ void tdm_load(u32x4 g0, i32x8 g1, i32x4 g2, i32x4 g3) {
#if defined(__clang_major__) && (__clang_major__ >= 23)
    i32x8 z8 = {};
    __builtin_amdgcn_tensor_load_to_lds(g0, g1, g2, g3, z8, 0);
#else
    __builtin_amdgcn_tensor_load_to_lds(g0, g1, g2, g3, 0);
#endif
}

// group0: [1:0]=count=1, [31]=gather_mode, [63:32]=lds_addr,
//         [120:64]=global_addr, [127:126]=type=2
__device__ __forceinline__ u32x4 tdm_group0(unsigned lds_addr, const void* gptr,
                                            bool gather) {
    unsigned long long ga = (unsigned long long)(uintptr_t)gptr;
    u32x4 g0;
    g0.x = 1u | (gather ? (1u << 31) : 0u);
    g0.y = lds_addr;
    g0.z = (unsigned)(ga & 0xFFFFFFFFu);
    g0.w = (unsigned)((ga >> 32) & 0x01FFFFFFu) | (2u << 30);
    return g0;
}

// group1: data_size=4B, pad_enable, pad_interval=7 (256 dw), pad_amount=1 (2 dw),
//         tensor_dim0/1, tile_dim0/1 (tile_dim2=0), tensor_dim0_stride.
__device__ __forceinline__ i32x8 tdm_group1(unsigned tdim0, unsigned tdim1,
                                            unsigned tile0, unsigned tile1,
                                            unsigned stride0) {
    i32x8 g;
    g[0] = (2 << 16) | (1 << 20) | (7 << 22) | (1 << 25);
    g[1] = (int)((tdim0 & 0xFFFFu) << 16);                               // [63:48]=dim0 lo
    g[2] = (int)(((tdim0 >> 16) & 0xFFFFu) | ((tdim1 & 0xFFFFu) << 16)); // dim0 hi | dim1 lo
    g[3] = (int)(((tdim1 >> 16) & 0xFFFFu) | ((tile0 & 0xFFFFu) << 16)); // dim1 hi | tile0
    g[4] = (int)(tile1 & 0xFFFFu);                                       // tile1 | tile2=0
    g[5] = (int)stride0;                                                 // stride0[31:0]
    g[6] = 0;                                                            // stride0 hi | stride1 lo
    g[7] = 0;
    return g;
}

// Gather-mode load: 16-bit row indices (token ids) in groups 2/3 pull `nrows`
// rows of x[T,512] straight into LDS (stride-516 padded layout).
__device__ __forceinline__ void tdm_gather_x(float* sX, const float* x,
                                             const int* tl, int base, int nrows) {
    unsigned idx[16];
    const int4* p = (const int4*)(tl + base);   // 64B-aligned (base % 16 == 0)
#pragma unroll
    for (int j = 0; j < 4; ++j) {
        int4 v = p[j];
        idx[j * 4 + 0] = (unsigned)v.x; idx[j * 4 + 1] = (unsigned)v.y;
        idx[j * 4 + 2] = (unsigned)v.z; idx[j * 4 + 3] = (unsigned)v.w;
    }
    i32x4 g2, g3;
#pragma unroll
    for (int j = 0; j < 4; ++j) {
        g2[j] = (int)((idx[2 * j] & 0xFFFFu)     | (idx[2 * j + 1] << 16));
        g3[j] = (int)((idx[8 + 2 * j] & 0xFFFFu) | (idx[8 + 2 * j + 1] << 16));
    }
    u32x4 g0 = tdm_group0((unsigned)(uintptr_t)sX, x, true);
    i32x8 g1 = tdm_group1(D_DIM, T_TOK, D_DIM, (unsigned)nrows, D_DIM);
    tdm_load(g0, g1, g2, g3);
}

// ---------------------------------------------------------------------------
// Kernel 1: gating logits + top-2 + softmax(top-2). One thread per token.
// ---------------------------------------------------------------------------
__global__ __launch_bounds__(256) void gate_topk_kernel(
    const float* __restrict__ x, const float* __restrict__ gw,
    int* __restrict__ top_idx, float* __restrict__ routing_w)
{
    int t = blockIdx.x * blockDim.x + threadIdx.x;
    if (t >= T_TOK) return;
    const float* xr = x + (size_t)t * D_DIM;

    float acc[E_EXP];
#pragma unroll
    for (int e = 0; e < E_EXP; ++e) acc[e] = 0.f;

    for (int d = 0; d < D_DIM; d += 4) {
        float4 xv = *(const float4*)(xr + d);
#pragma unroll
        for (int e = 0; e < E_EXP; ++e) {
            const float* g = gw + e * D_DIM + d;  // uniform across wave
            acc[e] += xv.x * g[0] + xv.y * g[1] + xv.z * g[2] + xv.w * g[3];
        }
    }
    int i0 = 0; float v0 = acc[0];
#pragma unroll
    for (int e = 1; e < E_EXP; ++e) if (acc[e] > v0) { v0 = acc[e]; i0 = e; }
    int i1 = -1; float v1 = -__builtin_inff();
#pragma unroll
    for (int e = 0; e < E_EXP; ++e) if (e != i0 && acc[e] > v1) { v1 = acc[e]; i1 = e; }

    float w0 = 1.f / (1.f + __expf(v1 - v0));   // softmax over {v0, v1}
    top_idx[t * 2 + 0] = i0;
    top_idx[t * 2 + 1] = i1;
    routing_w[t * 2 + 0] = w0;
    routing_w[t * 2 + 1] = 1.f - w0;
}

// ---------------------------------------------------------------------------
// Kernel 2: deterministic ordered per-expert token-list compaction (wave32
// ballot-prefix; no atomics -> bit-deterministic downstream reductions).
// ---------------------------------------------------------------------------
__global__ __launch_bounds__(32) void route_compact_kernel(
    const int* __restrict__ top_idx, int* __restrict__ counts,
    int* __restrict__ token_list)
{
    const int e = blockIdx.x;
    const int lane = threadIdx.x;
    int cnt = 0;
    for (int base = 0; base < T_TOK; base += 32) {
        int t = base + lane;
        bool flag = (top_idx[t * 2] == e) || (top_idx[t * 2 + 1] == e);
        unsigned mask = (unsigned)__ballot(flag);
        int prefix = __popc(mask & ((1u << lane) - 1u));
        if (flag) token_list[e * T_TOK + cnt + prefix] = t;
        cnt += __popc(mask);
    }
    if (lane == 0) counts[e] = cnt;
}

// ---------------------------------------------------------------------------
// Kernel 3: h_mean[e,f] = mean_t silu(x_t . w1_e[f,:]) over routed tokens.
// Grid (F/128, E), 256 thr = 8 waves. TDM stages the 128x512 w1 tile (one
// tensor_load_to_lds, padded rows) and gathers 16 token rows per chunk
// (gather-mode tensor_load_to_lds). Waves do only ds_load -> v_wmma.
// ---------------------------------------------------------------------------
__global__ __launch_bounds__(256) void expert_h_kernel(
    const float* __restrict__ x, const float* __restrict__ w1,
    const int* __restrict__ counts, const int* __restrict__ token_list,
    float* __restrict__ h_mean)
{
    extern __shared__ float smem[];            // [128*516] w1  +  [16*516] x
    float* sW = smem;
    float* sX = smem + TILE_F * W1_STRIDE;

    const int e    = blockIdx.y;
    const int f0   = blockIdx.x * TILE_F;
    const int tid  = threadIdx.x;
    const int wv   = tid >> 5;
    const int lane = tid & 31;
    const int half = lane >> 4;
    const int ln   = lane & 15;
    const int ko   = 2 * half;                 // K pair select (A/B f32 layout)

    const int  ne     = counts[e];
    const int  nchunk = (ne + 15) >> 4;
    const int* tl     = token_list + e * T_TOK;

    if (tid == 0) {
        // w1[e, f0:f0+128, :] -> LDS, one DMA (rows padded to stride 516)
        u32x4 g0 = tdm_group0((unsigned)(uintptr_t)sW,
                              w1 + ((size_t)e * F_DIM + f0) * D_DIM, false);
        i32x8 g1 = tdm_group1(D_DIM, TILE_F, D_DIM, TILE_F, D_DIM);
        i32x4 z  = {};
        tdm_load(g0, g1, z, z);
        if (nchunk > 0)
            tdm_gather_x(sX, x, tl, 0, ne < 16 ? ne : 16);
    }
    __builtin_amdgcn_s_wait_tensorcnt(0);
    __syncthreads();

    float partial = 0.f;                       // per-lane F-column partial sum
    const float* aBase = sX + ln * W1_STRIDE;               // A: M = ln
    const float* bBase = sW + (wv * 16 + ln) * W1_STRIDE;   // B: N = ln

    for (int c = 0; c < nchunk; ++c) {
        const int base = c * 16;
        v8f acc0 = {}, acc1 = {};
        // K = 512 as two 256-dword segments (TDM pad sits between them)
#pragma unroll
        for (int seg = 0; seg < 2; ++seg) {
            const float* aSeg = aBase + seg * SEG_STRIDE;
            const float* bSeg = bBase + seg * SEG_STRIDE;
#pragma unroll 4
            for (int k = 0; k < 256; k += 8) {
                v2f a0 = *(const v2f*)(aSeg + k + ko);
                v2f b0 = *(const v2f*)(bSeg + k + ko);
                acc0 = __builtin_amdgcn_wmma_f32_16x16x4_f32(
                    false, a0, false, b0, (short)0, acc0, false, false);
                v2f a1 = *(const v2f*)(aSeg + k + 4 + ko);
                v2f b1 = *(const v2f*)(bSeg + k + 4 + ko);
                acc1 = __builtin_amdgcn_wmma_f32_16x16x4_f32(
                    false, a1, false, b1, (short)0, acc1, false, false);
            }
        }
        // silu + validity-masked row-sum. C/D layout: VGPR r, half h -> M=r+8h.
#pragma unroll
        for (int r = 0; r < 8; ++r) {
            int m = r + 8 * half;
            float v = acc0[r] + acc1[r];
            float s = v * __builtin_amdgcn_rcpf(1.f + __expf(-v));  // v_rcp_f32
            partial += ((base + m) < ne) ? s : 0.f;
        }
        __syncthreads();                       // everyone done reading sX
        if (c + 1 < nchunk) {
            if (tid == 0) {
                int nb = ne - (c + 1) * 16;
                tdm_gather_x(sX, x, tl, (c + 1) * 16, nb < 16 ? nb : 16);
            }
            __builtin_amdgcn_s_wait_tensorcnt(0);
        }
        __syncthreads();                       // fresh sX visible to all
    }
    // fold M-halves: lanes n and n+16 hold partials for the same N=n
    partial += __shfl_xor(partial, 16, 32);
    if (lane < 16) {
        float invc = 1.f / (float)(ne > 0 ? ne : 1);
        h_mean[(size_t)e * F_DIM + f0 + wv * 16 + lane] = partial * invc;
    }
}

// ---------------------------------------------------------------------------
// Kernel 4: y_mean[e,d] = h_mean[e,:] . w2[e,d,:].  One wave per d.
// ---------------------------------------------------------------------------
__global__ __launch_bounds__(256) void ymean_kernel(
    const float* __restrict__ h_mean, const float* __restrict__ w2,
    float* __restrict__ y_mean)
{
    const int e    = blockIdx.y;
    const int d    = blockIdx.x * 8 + (threadIdx.x >> 5);
    const int lane = threadIdx.x & 31;
    const float* h = h_mean + (size_t)e * F_DIM;
    const float* w = w2 + ((size_t)e * D_DIM + d) * F_DIM;

    float sum = 0.f;
    for (int f4 = lane; f4 < F_DIM / 4; f4 += 32) {
        float4 hv = *(const float4*)(h + f4 * 4);
        float4 wv = *(const float4*)(w + f4 * 4);
        sum += hv.x * wv.x + hv.y * wv.y + hv.z * wv.z + hv.w * wv.w;
    }
#pragma unroll
    for (int off = 16; off; off >>= 1) sum += __shfl_down(sum, off, 32);
    if (lane == 0) y_mean[(size_t)e * D_DIM + d] = sum;
}

// ---------------------------------------------------------------------------
// Kernel 5: out[t,d] = sum_k routing_w[t,k] * y_mean[top_idx[t,k], d]
// ---------------------------------------------------------------------------
__global__ __launch_bounds__(256) void combine_kernel(
    const int* __restrict__ top_idx, const float* __restrict__ routing_w,
    const float* __restrict__ y_mean, float* __restrict__ out)
{
    int i = blockIdx.x * 256 + threadIdx.x;
    if (i >= T_TOK * D_DIM) return;
    int t = i >> 9;
    int d = i & (D_DIM - 1);
    int  i0 = top_idx[t * 2],     i1 = top_idx[t * 2 + 1];
    float w0 = routing_w[t * 2],  w1 = routing_w[t * 2 + 1];
    out[i] = w0 * y_mean[i0 * D_DIM + d] + w1 * y_mean[i1 * D_DIM + d];
}

// ---------------------------------------------------------------------------
extern "C" void kernel_launch(void* const* d_in, const int* in_sizes, int n_in,
                              void* d_out, int out_size, void* d_ws, size_t ws_size,
                              hipStream_t stream) {
    const float* x  = (const float*)d_in[0];   // [4,512,512]
    const float* gw = (const float*)d_in[1];   // [16,512]
    const float* w1 = (const float*)d_in[2];   // [16,2048,512]
    const float* w2 = (const float*)d_in[3];   // [16,512,2048]
    float* out = (float*)d_out;                // [4,512,512]

    char* ws = (char*)d_ws;
    int*   top_idx    = (int*)ws;    ws += (size_t)T_TOK * 2 * sizeof(int);
    float* routing_w  = (float*)ws;  ws += (size_t)T_TOK * 2 * sizeof(float);
    int*   counts     = (int*)ws;    ws += 256;
    int*   token_list = (int*)ws;    ws += (size_t)(E_EXP * T_TOK + 64) * sizeof(int);
    float* h_mean     = (float*)ws;  ws += (size_t)E_EXP * F_DIM * sizeof(float);
    float* y_mean     = (float*)ws;  ws += (size_t)E_EXP * D_DIM * sizeof(float);

    gate_topk_kernel<<<T_TOK / 256, 256, 0, stream>>>(x, gw, top_idx, routing_w);
    route_compact_kernel<<<E_EXP, 32, 0, stream>>>(top_idx, counts, token_list);

    size_t lds_bytes = (size_t)(TILE_F + 16) * W1_STRIDE * sizeof(float); // ~291 KB
    dim3 g2(F_DIM / TILE_F, E_EXP);
    expert_h_kernel<<<g2, 256, lds_bytes, stream>>>(x, w1, counts, token_list, h_mean);

    dim3 g3(D_DIM / 8, E_EXP);
    ymean_kernel<<<g3, 256, 0, stream>>>(h_mean, w2, y_mean);

    combine_kernel<<<(T_TOK * D_DIM) / 256, 256, 0, stream>>>(top_idx, routing_w, y_mean, out);
}